// MemoryBank_41832981463304
// MI455X (gfx1250) — compile-verified
//
#include <hip/hip_runtime.h>
#include <hip/hip_bf16.h>
#include <math.h>

typedef float v2f __attribute__((ext_vector_type(2)));
typedef float v8f __attribute__((ext_vector_type(8)));

#define DD 128      // feature dim
#define KT 32       // keys per LDS tile
#define WAVES 8
#define KPITCH 132  // LDS pitch for key tile (bank-conflict padding, 16B aligned)
#define PPITCH 36   // LDS pitch for P scratch

#if __has_builtin(__builtin_amdgcn_global_load_async_to_lds_b128)
#define HAVE_ASYNC 1
// Pointer types matching the builtin's signature (from the round-2 diagnostic):
// param0 = vector_size(16) int, global (AS1); param1 = same, LDS (AS3).
typedef int vsi4 __attribute__((vector_size(16)));
typedef vsi4 __attribute__((address_space(1))) * as1_v4i;
typedef vsi4 __attribute__((address_space(3))) * as3_v4i;
#else
#define HAVE_ASYNC 0
#endif

template <int N>
static __device__ __forceinline__ void wait_async() {
#if HAVE_ASYNC
#if __has_builtin(__builtin_amdgcn_s_wait_asynccnt)
  __builtin_amdgcn_s_wait_asynccnt((unsigned short)N);
#else
  asm volatile("s_wait_asynccnt %0" ::"i"(N) : "memory");
#endif
#endif
}

static __device__ __forceinline__ v8f wmma_f32(v2f a, v2f b, v8f c) {
  // (neg_a, A, neg_b, B, c_mod, C, reuse_a, reuse_b) -> v_wmma_f32_16x16x4_f32
  return __builtin_amdgcn_wmma_f32_16x16x4_f32(false, a, false, b, (short)0, c,
                                               false, false);
}

// Stage one 32x128 f32 key tile into LDS (async if toolchain supports it).
static __device__ __forceinline__ void issue_tile(float dst[KT][KPITCH],
                                                  const float* __restrict__ keys,
                                                  int kb, int tid, int Ktot) {
  const int c4 = tid & 31;
  const int r0 = tid >> 5;
#pragma unroll
  for (int i = 0; i < 4; ++i) {
    const int row = i * 8 + r0;
    const float* g = keys + (size_t)(kb + row) * DD + c4 * 4;
    float* l = &dst[row][c4 * 4];
#if HAVE_ASYNC
    __builtin_amdgcn_global_load_async_to_lds_b128((as1_v4i)g, (as3_v4i)l, 0, 0);
#else
    *(float4*)l = *(const float4*)g;
    if (kb + KT < Ktot) __builtin_prefetch(g + (size_t)KT * DD, 0, 0);
#endif
  }
}

// Pass 1: per-key inverse norms (values stay raw; only scores use normalized keys)
__global__ void keynorm_kernel(const float* __restrict__ keys,
                               float* __restrict__ invk, int K) {
  int k = blockIdx.x * blockDim.x + threadIdx.x;
  if (k >= K) return;
  const float4* row = (const float4*)(keys + (size_t)k * DD);
  float s = 0.f;
#pragma unroll
  for (int i = 0; i < DD / 4; ++i) {
    float4 v = row[i];
    s += v.x * v.x + v.y * v.y + v.z * v.z + v.w * v.w;
  }
  invk[k] = 1.0f / fmaxf(sqrtf(s), 1e-12f);
}

// Pass 2: flash-attention style retrieve. 8 waves/block, 16 query rows/wave.
__global__ __launch_bounds__(256) void retrieve_kernel(
    const float* __restrict__ q, const float* __restrict__ keys,
    const float* __restrict__ invk, float* __restrict__ out, int Ktot) {
  __shared__ float kt[2][KT][KPITCH];        // double-buffered raw key tiles
  __shared__ float pbuf[WAVES][16][PPITCH];  // per-wave P staging (C->A relayout)

  const int tid = threadIdx.x;
  const int wave = tid >> 5;
  const int lane = tid & 31;
  const int hi = lane >> 4;  // 0: lanes 0-15, 1: lanes 16-31
  const int l16 = lane & 15;

  // ---- load 16 query rows as fp32 A-fragments, L2-normalize in registers ----
  // A 16x4 layout: lane holds row M=l16; K = v + 2*hi (per 4-wide K step)
  const size_t qrow = (size_t)blockIdx.x * 128 + (size_t)wave * 16 + l16;
  const float4* q4 = (const float4*)(q + qrow * DD);
  v2f qf[32];
  float ss = 0.f;
#pragma unroll
  for (int i = 0; i < 32; ++i) {
    float4 v = q4[i];
    ss += v.x * v.x + v.y * v.y + v.z * v.z + v.w * v.w;
    v2f t;
    if (hi) { t.x = v.z; t.y = v.w; } else { t.x = v.x; t.y = v.y; }
    qf[i] = t;
  }
  const float invq = 1.0f / fmaxf(sqrtf(ss), 1e-12f);
#pragma unroll
  for (int i = 0; i < 32; ++i) qf[i] *= invq;

  const v8f zero = {0.f, 0.f, 0.f, 0.f, 0.f, 0.f, 0.f, 0.f};
  v8f O[8];  // 16x128 output tile: O[c][r] -> row M=r+8*hi, col d=c*16+l16
#pragma unroll
  for (int c = 0; c < 8; ++c) O[c] = zero;

  float m[8], lsum[8];
#pragma unroll
  for (int r = 0; r < 8; ++r) {
    m[r] = -__builtin_inff();
    lsum[r] = 0.f;
  }

  // one key-tile worth of compute: GEMM1 + online softmax + GEMM2
  auto compute_tile = [&](const float(*ktile)[KPITCH], int kb0) {
    const float ik0 = invk[kb0 + l16];
    const float ik1 = invk[kb0 + 16 + l16];

    // ---- GEMM1: S(16x32) = q_hat(16x128) . K_tile^T ----
    v8f S0 = zero, S1 = zero;
#pragma unroll
    for (int kk = 0; kk < 32; ++kk) {
      const int dbase = 4 * kk + 2 * hi;  // B 4x16 layout: k = v + 2*hi
      v2f b0 = *(const v2f*)(&ktile[l16][dbase]);
      v2f b1 = *(const v2f*)(&ktile[16 + l16][dbase]);
      S0 = wmma_f32(qf[kk], b0, S0);
      S1 = wmma_f32(qf[kk], b1, S1);
    }
    S0 *= ik0;  // fold key normalization into scores (beta == 1)
    S1 *= ik1;

    // ---- online softmax (row M = r + 8*hi lives in 16 lanes of one half) ----
    float alpha[8];
#pragma unroll
    for (int r = 0; r < 8; ++r) {
      float mr = fmaxf(S0[r], S1[r]);
      mr = fmaxf(mr, __shfl_xor(mr, 1, 32));
      mr = fmaxf(mr, __shfl_xor(mr, 2, 32));
      mr = fmaxf(mr, __shfl_xor(mr, 4, 32));
      mr = fmaxf(mr, __shfl_xor(mr, 8, 32));
      const float mn = fmaxf(m[r], mr);
      alpha[r] = __expf(m[r] - mn);
      m[r] = mn;
      const float p0 = __expf(S0[r] - mn);
      const float p1 = __expf(S1[r] - mn);
      S0[r] = p0;
      S1[r] = p1;
      float rs = p0 + p1;
      rs += __shfl_xor(rs, 1, 32);
      rs += __shfl_xor(rs, 2, 32);
      rs += __shfl_xor(rs, 4, 32);
      rs += __shfl_xor(rs, 8, 32);
      lsum[r] = lsum[r] * alpha[r] + rs;
    }
#pragma unroll
    for (int c = 0; c < 8; ++c)
#pragma unroll
      for (int r = 0; r < 8; ++r) O[c][r] *= alpha[r];

    // ---- relayout P: C-layout -> A-layout via per-wave LDS scratch ----
#pragma unroll
    for (int r = 0; r < 8; ++r) {
      pbuf[wave][r + 8 * hi][l16] = S0[r];
      pbuf[wave][r + 8 * hi][16 + l16] = S1[r];
    }
    v2f af[8];
#pragma unroll
    for (int kk = 0; kk < 8; ++kk)
      af[kk] = *(const v2f*)(&pbuf[wave][l16][4 * kk + 2 * hi]);

    // ---- GEMM2: O(16x128) += P(16x32) . K_tile(32x128) ----
#pragma unroll
    for (int c = 0; c < 8; ++c) {
#pragma unroll
      for (int kk = 0; kk < 8; ++kk) {
        const int kl = 4 * kk + 2 * hi;
        v2f b;
        b.x = ktile[kl][c * 16 + l16];
        b.y = ktile[kl + 1][c * 16 + l16];
        O[c] = wmma_f32(af[kk], b, O[c]);
      }
    }
  };

  // ---- double-buffered, async-staged key loop (Ktot % (2*KT) == 0) ----
  issue_tile(kt[0], keys, 0, tid, Ktot);
  for (int kb = 0; kb + 2 * KT <= Ktot; kb += 2 * KT) {
    issue_tile(kt[1], keys, kb + KT, tid, Ktot);
    wait_async<4>();  // tile in kt[0] complete; kt[1] may be in flight
    __syncthreads();
    compute_tile(kt[0], kb);
    __syncthreads();

    if (kb + 2 * KT < Ktot) {
      issue_tile(kt[0], keys, kb + 2 * KT, tid, Ktot);
      wait_async<4>();
    } else {
      wait_async<0>();
    }
    __syncthreads();
    compute_tile(kt[1], kb + KT);
    __syncthreads();
  }

  // ---- epilogue: divide by softmax denominator, coalesced store ----
  float linv[8];
#pragma unroll
  for (int r = 0; r < 8; ++r) linv[r] = 1.0f / lsum[r];
  const size_t orow = (size_t)blockIdx.x * 128 + (size_t)wave * 16;
#pragma unroll
  for (int c = 0; c < 8; ++c)
#pragma unroll
    for (int r = 0; r < 8; ++r)
      out[(orow + r + 8 * hi) * DD + c * 16 + l16] = O[c][r] * linv[r];
}

extern "C" void kernel_launch(void* const* d_in, const int* in_sizes, int n_in,
                              void* d_out, int out_size, void* d_ws,
                              size_t ws_size, hipStream_t stream) {
  const float* q = (const float*)d_in[0];
  const float* keys = (const float*)d_in[1];
  float* out = (float*)d_out;
  float* invk = (float*)d_ws;  // K floats of scratch
  const int N = in_sizes[0] / DD;
  const int K = in_sizes[1] / DD;
  keynorm_kernel<<<(K + 255) / 256, 256, 0, stream>>>(keys, invk, K);
  retrieve_kernel<<<N / 128, 256, 0, stream>>>(q, keys, invk, out, K);
}